// Ssim_grad_L1_5196910428171
// MI455X (gfx1250) — compile-verified
//
#include <hip/hip_runtime.h>

// ---------------------------------------------------------------------------
// SSIM + gradient-L1 + depth-L1 loss for (32,1,480,640) f32 images, MI455X.
// SSIM's separable 11-tap Gaussian convs run on the matrix pipe via
// V_WMMA_F32_16X16X4_F32 (FP32 WMMA, 16x16 output tiles, K chained 7x4=28).
// Tile staging uses CDNA5 async global->LDS copies (ASYNCcnt-tracked).
// ---------------------------------------------------------------------------

typedef float v2f __attribute__((ext_vector_type(2)));
typedef float v8f __attribute__((ext_vector_type(8)));

constexpr int IMG_H = 480;
constexpr int IMG_W = 640;
constexpr int NBATCH = 32;
constexpr int OUT_H = 470;          // 480 - 11 + 1
constexpr int OUT_W = 630;          // 640 - 11 + 1
constexpr int TILES_Y = 30;         // ceil(470/16)
constexpr int TILES_X = 40;         // ceil(630/16)
constexpr int WAVES_PER_BLOCK = 4;

// Normalized 1D Gaussian, sigma=1.5, 11 taps (matches reference window).
__constant__ float GW[11] = {
    0.00102838f, 0.00759878f, 0.03600077f, 0.10936010f, 0.21300553f,
    0.26601171f,
    0.21300553f, 0.10936010f, 0.03600077f, 0.00759878f, 0.00102838f};

__device__ __forceinline__ v8f wmma4(v2f a, v2f b, v8f c) {
  // D(16x16 f32) = A(16x4 f32) * B(4x16 f32) + C
  return __builtin_amdgcn_wmma_f32_16x16x4_f32(
      /*neg_a=*/false, a, /*neg_b=*/false, b,
      /*c_mod=*/(short)0, c, /*reuse_a=*/false, /*reuse_b=*/false);
}

// CDNA5 async global->LDS B32 copy, GV mode (64-bit vaddr, no SADDR).
// Copies 4 bytes from the global address in vaddr to the LDS address in the first operand; tracked with ASYNCcnt, honors EXEC per lane.
__device__ __forceinline__ void async_load_b32(unsigned lds_byte_addr,
                                               const float* gaddr) {
  asm volatile("global_load_async_to_lds_b32 %0, %1, off"
               :
               : "v"(lds_byte_addr),
                 "v"((unsigned long long)(uintptr_t)gaddr)
               : "memory");
}

__device__ __forceinline__ void wait_asynccnt0() {
  asm volatile("s_wait_asynccnt 0x0" ::: "memory");
}

__device__ __forceinline__ float wave_reduce_add(float v) {
  v += __shfl_xor(v, 16, 32);
  v += __shfl_xor(v, 8, 32);
  v += __shfl_xor(v, 4, 32);
  v += __shfl_xor(v, 2, 32);
  v += __shfl_xor(v, 1, 32);
  return v;
}

// ---------------------------------------------------------------------------
// ws[0]=sum|pred-true| ; ws[1]=sum(|grad_h diff|+|grad_w diff|) ; ws[2]=sum ssim
// ---------------------------------------------------------------------------
__global__ void init_ws_kernel(float* __restrict__ ws) {
  if (threadIdx.x < 4) ws[threadIdx.x] = 0.0f;
}

// Elementwise pass: depth L1 + np.gradient edge L1 (bandwidth-bound).
__global__ void edge_depth_kernel(const float* __restrict__ yt,
                                  const float* __restrict__ yp,
                                  float* __restrict__ ws) {
  const long total = (long)NBATCH * IMG_H * IMG_W;
  const long stride = (long)gridDim.x * blockDim.x;
  float sd = 0.0f, se = 0.0f;
  for (long i = (long)blockIdx.x * blockDim.x + threadIdx.x; i < total;
       i += stride) {
    if (i + stride < total) {  // speculative prefetch -> global_prefetch_b8
      __builtin_prefetch(yp + i + stride, 0, 1);
      __builtin_prefetch(yt + i + stride, 0, 1);
    }
    const int w = (int)(i % IMG_W);
    const int h = (int)((i / IMG_W) % IMG_H);
    const float c = yp[i] - yt[i];
    sd += fabsf(c);
    float gh, gw;
    if (h == 0) {
      gh = (yp[i + IMG_W] - yt[i + IMG_W]) - c;
    } else if (h == IMG_H - 1) {
      gh = c - (yp[i - IMG_W] - yt[i - IMG_W]);
    } else {
      gh = 0.5f * ((yp[i + IMG_W] - yt[i + IMG_W]) -
                   (yp[i - IMG_W] - yt[i - IMG_W]));
    }
    if (w == 0) {
      gw = (yp[i + 1] - yt[i + 1]) - c;
    } else if (w == IMG_W - 1) {
      gw = c - (yp[i - 1] - yt[i - 1]);
    } else {
      gw = 0.5f * ((yp[i + 1] - yt[i + 1]) - (yp[i - 1] - yt[i - 1]));
    }
    se += fabsf(gh) + fabsf(gw);
  }
  sd = wave_reduce_add(sd);
  se = wave_reduce_add(se);
  if ((threadIdx.x & 31) == 0) {
    atomicAdd(&ws[0], sd);
    atomicAdd(&ws[1], se);
  }
}

// ---------------------------------------------------------------------------
// SSIM tile kernel: one wave per 16x16 output tile.
//  horizontal: T(32x16) = I(32x28) x G(28x16), G[j][n] = g[j-n] (banded)
//  vertical:   O(16x16) = Gt(16x28) x T(28x16)
// A 16x4 f32 lane layout: K = 2*(lane>>4) + vgpr ; B 4x16 identical formula,
// so ONE banded fragment Wc[k] serves as horizontal-B and vertical-A.
// ---------------------------------------------------------------------------
__global__ __launch_bounds__(WAVES_PER_BLOCK * 32) void ssim_tile_kernel(
    const float* __restrict__ img1, const float* __restrict__ img2,
    float* __restrict__ ws) {
  __shared__ float lA[WAVES_PER_BLOCK][32 * 28];
  __shared__ float lB[WAVES_PER_BLOCK][32 * 28];
  __shared__ float lT[WAVES_PER_BLOCK][32 * 16];
  __shared__ float lg[16];

  if (threadIdx.x < 11) lg[threadIdx.x] = GW[threadIdx.x];
  __syncthreads();

  const int wave = threadIdx.x >> 5;
  const int lane = threadIdx.x & 31;
  const int n = lane & 15;   // output col (B/C/D) and matrix row m (A)
  const int hi = lane >> 4;  // lane-half selector

  const long tileId = (long)blockIdx.x * WAVES_PER_BLOCK + wave;
  const long totalTiles = (long)NBATCH * TILES_Y * TILES_X;
  if (tileId >= totalTiles) return;

  const int img = (int)(tileId / (TILES_Y * TILES_X));
  const int t = (int)(tileId % (TILES_Y * TILES_X));
  const int r0 = (t / TILES_X) * 16;
  const int c0 = (t % TILES_X) * 16;

  const size_t base = (size_t)img * (IMG_H * IMG_W);
  float* lin1 = lA[wave];
  float* lin2 = lB[wave];
  float* lt = lT[wave];

  // Stage clamped 32x28 input window straight into LDS with CDNA5 async
  // copies (lanes 0..27 = columns; per-lane EXEC masks the rest). The
  // VGPR file is untouched during the transfer.
  if (lane < 28) {
    int gc = c0 + lane;
    if (gc > IMG_W - 1) gc = IMG_W - 1;
#pragma unroll 4
    for (int r = 0; r < 32; ++r) {
      int gr = r0 + r;
      if (gr > IMG_H - 1) gr = IMG_H - 1;
      const size_t off = base + (size_t)gr * IMG_W + gc;
      async_load_b32((unsigned)(uintptr_t)&lin1[r * 28 + lane], img1 + off);
      async_load_b32((unsigned)(uintptr_t)&lin2[r * 28 + lane], img2 + off);
    }
  }

  // Banded Gaussian operand fragments (K index = 4k + 2*hi + vgpr);
  // computed while the async transfers are in flight.
  v2f Wc[7];
#pragma unroll
  for (int k = 0; k < 7; ++k) {
    const int j0 = 4 * k + 2 * hi;
    const int d0 = j0 - n, d1 = j0 + 1 - n;
    Wc[k].x = (d0 >= 0 && d0 <= 10) ? lg[d0] : 0.0f;
    Wc[k].y = (d1 >= 0 && d1 <= 10) ? lg[d1] : 0.0f;
  }

  // Async deposits are tracked by ASYNCcnt (not DScnt): drain before reading.
  wait_asynccnt0();

  v8f O[5];
#pragma unroll
  for (int s = 0; s < 5; ++s) {
    // Horizontal pass: two 16-row blocks, K=28 chained as 7 x WMMA(K=4).
#pragma unroll
    for (int rb = 0; rb < 2; ++rb) {
      v8f acc = {};
#pragma unroll
      for (int k = 0; k < 7; ++k) {
        const int idx = (rb * 16 + n) * 28 + 4 * k + 2 * hi;
        const float a1 = lin1[idx], b1 = lin1[idx + 1];
        const float a2 = lin2[idx], b2 = lin2[idx + 1];
        v2f A;
        if (s == 0) {
          A.x = a1; A.y = b1;                 // mu1 source
        } else if (s == 1) {
          A.x = a2; A.y = b2;                 // mu2 source
        } else if (s == 2) {
          A.x = a1 * a1; A.y = b1 * b1;       // E[x^2]
        } else if (s == 3) {
          A.x = a2 * a2; A.y = b2 * b2;       // E[y^2]
        } else {
          A.x = a1 * a2; A.y = b1 * b2;       // E[xy]
        }
        acc = wmma4(A, Wc[k], acc);
      }
      // Spill T (C/D layout: vgpr v -> rows v and v+8) to the wave-private
      // LDS slice; wave-internal DS ordering makes this safe without barriers.
#pragma unroll
      for (int v = 0; v < 8; ++v)
        lt[(rb * 16 + v + 8 * hi) * 16 + n] = acc[v];
    }
    // Vertical pass: O = Gt x T. Gt columns j>=26 are structurally zero, so
    // K padded to 28 is exact.
    v8f accO = {};
#pragma unroll
    for (int k = 0; k < 7; ++k) {
      const int j = 4 * k + 2 * hi;
      v2f Bv;
      Bv.x = lt[j * 16 + n];
      Bv.y = lt[(j + 1) * 16 + n];
      accO = wmma4(Wc[k], Bv, accO);
    }
    O[s] = accO;
  }

  // ssim_map from the five local statistics; mask to the 470x630 valid region.
  const float C1 = 1.0f;  // (0.01*100)^2
  const float C2 = 9.0f;  // (0.03*100)^2
  float part = 0.0f;
#pragma unroll
  for (int v = 0; v < 8; ++v) {
    const float mu1 = O[0][v], mu2 = O[1][v];
    const float mu1s = mu1 * mu1, mu2s = mu2 * mu2, m12 = mu1 * mu2;
    const float s1 = O[2][v] - mu1s;
    const float s2 = O[3][v] - mu2s;
    const float s12 = O[4][v] - m12;
    const float num = (2.0f * m12 + C1) * (2.0f * s12 + C2);
    const float den = (mu1s + mu2s + C1) * (s1 + s2 + C2);
    const int row = r0 + v + 8 * hi;
    const int col = c0 + n;
    if (row < OUT_H && col < OUT_W) part += num / den;
  }
  part = wave_reduce_add(part);
  if (lane == 0) atomicAdd(&ws[2], part);
}

__global__ void finalize_kernel(const float* __restrict__ ws,
                                float* __restrict__ out) {
  if (threadIdx.x == 0 && blockIdx.x == 0) {
    const float NT = (float)NBATCH * IMG_H * IMG_W;   // 9,830,400
    const float NS = (float)NBATCH * OUT_H * OUT_W;   // 9,475,200
    const float ssim_mean = ws[2] / NS;
    float l_ssim = (1.0f - ssim_mean) * 0.5f;
    l_ssim = fminf(fmaxf(l_ssim, 0.0f), 1.0f);
    out[0] = l_ssim + ws[1] / NT + 0.1f * (ws[0] / NT);
  }
}

extern "C" void kernel_launch(void* const* d_in, const int* in_sizes, int n_in,
                              void* d_out, int out_size, void* d_ws,
                              size_t ws_size, hipStream_t stream) {
  (void)in_sizes; (void)n_in; (void)out_size; (void)ws_size;
  const float* y_true = (const float*)d_in[0];
  const float* y_pred = (const float*)d_in[1];
  float* out = (float*)d_out;
  float* ws = (float*)d_ws;

  hipLaunchKernelGGL(init_ws_kernel, dim3(1), dim3(32), 0, stream, ws);
  hipLaunchKernelGGL(edge_depth_kernel, dim3(2048), dim3(256), 0, stream,
                     y_true, y_pred, ws);
  const long totalTiles = (long)NBATCH * TILES_Y * TILES_X;  // 38400
  const int blocks = (int)((totalTiles + WAVES_PER_BLOCK - 1) / WAVES_PER_BLOCK);
  hipLaunchKernelGGL(ssim_tile_kernel, dim3(blocks), dim3(WAVES_PER_BLOCK * 32),
                     0, stream, y_pred, y_true, ws);
  hipLaunchKernelGGL(finalize_kernel, dim3(1), dim3(1), 0, stream, ws, out);
}